// SBEpisodicMemory_28587302323145
// MI455X (gfx1250) — compile-verified
//
#include <hip/hip_runtime.h>
#include <math.h>

#define BB 64
#define DD 256
#define NN 2048

typedef float v2f __attribute__((ext_vector_type(2)));
typedef float v4f __attribute__((ext_vector_type(4)));
typedef float v8f __attribute__((ext_vector_type(8)));

__device__ __forceinline__ float sigm(float x) { return 1.f / (1.f + expf(-x)); }
__device__ __forceinline__ float d4(float4 a, float4 b) {
    return a.x*b.x + a.y*b.y + a.z*b.z + a.w*b.w;
}

// ---------------------------------------------------------------------------
// Kernel A: three 64x512x256 f32 matmuls via V_WMMA_F32_16X16X4_F32.
// which==0: [signal,previous] @ W_rq  -> rq_raw (pre-norm query)
// which==1: [signal,hidden]   @ W_key -> tanh(+b_key) -> ck
// which==2: [signal,hidden]   @ W_val -> tanh(+b_val) -> cv
// One wave per 16x16 output tile; K=512 in steps of 4.
// ---------------------------------------------------------------------------
__global__ __launch_bounds__(32) void mm_wmma(
    const float* __restrict__ signal, const float* __restrict__ previous,
    const float* __restrict__ hidden,
    const float* __restrict__ W_rq, const float* __restrict__ W_key,
    const float* __restrict__ b_key, const float* __restrict__ W_val,
    const float* __restrict__ b_val,
    float* __restrict__ rq_raw, float* __restrict__ ckO, float* __restrict__ cvO)
{
    const int nt = blockIdx.x;      // 16 col tiles
    const int mt = blockIdx.y;      // 4 row tiles
    const int which = blockIdx.z;   // 3 matmuls
    const int lane = threadIdx.x;
    const int mrow = mt * 16 + (lane & 15);
    const int ncol = nt * 16 + (lane & 15);
    const int khalf = (lane >> 4) << 1;   // 0 or 2

    const float* second = (which == 0) ? previous : hidden;
    const float* W = (which == 0) ? W_rq : ((which == 1) ? W_key : W_val);

    v8f acc = {0.f, 0.f, 0.f, 0.f, 0.f, 0.f, 0.f, 0.f};
    for (int k0 = 0; k0 < 512; k0 += 4) {
        const float* src = (k0 < 256) ? signal : second;  // uniform branch
        const int cc = (k0 & 255) + khalf;
        v2f a, bf;
        a.x  = src[mrow * 256 + cc];
        a.y  = src[mrow * 256 + cc + 1];
        bf.x = W[(k0 + khalf) * 256 + ncol];
        bf.y = W[(k0 + khalf + 1) * 256 + ncol];
        acc = __builtin_amdgcn_wmma_f32_16x16x4_f32(
            false, a, false, bf, (short)0, acc, false, false);
    }

    const int mbase = mt * 16 + ((lane >> 4) << 3);
    #pragma unroll
    for (int r = 0; r < 8; r++) {
        const int off = (mbase + r) * 256 + ncol;
        const float v = acc[r];
        if (which == 0)      rq_raw[off] = v;
        else if (which == 1) ckO[off] = tanhf(v + b_key[ncol]);
        else                 cvO[off] = tanhf(v + b_val[ncol]);
    }
}

// ---------------------------------------------------------------------------
// Kernel B: per-batch normalization of query & candidate key, scalar gates,
// and zeroing of content/persistent read accumulators.
// ---------------------------------------------------------------------------
__global__ __launch_bounds__(256) void prep(
    const float* __restrict__ signal, const float* __restrict__ previous,
    const float* __restrict__ hidden,
    const float* __restrict__ W_mix, const float* __restrict__ b_mix,
    const float* __restrict__ W_wg,  const float* __restrict__ b_wg,
    const float* __restrict__ W_pg,  const float* __restrict__ b_pg,
    const float* __restrict__ W_mg,  const float* __restrict__ b_mg,
    const float* __restrict__ rq_raw, const float* __restrict__ ckO,
    float* __restrict__ q, float* __restrict__ nck, float* __restrict__ gates,
    float* __restrict__ cread, float* __restrict__ pread)
{
    __shared__ float sm[256];
    const int b = blockIdx.x, t = threadIdx.x;

    auto bsum = [&](float v) -> float {
        sm[t] = v; __syncthreads();
        for (int s = 128; s > 0; s >>= 1) { if (t < s) sm[t] += sm[t + s]; __syncthreads(); }
        float r = sm[0]; __syncthreads();
        return r;
    };

    const float rq = rq_raw[b * 256 + t];
    float ss = bsum(rq * rq);
    q[b * 256 + t] = rq / fmaxf(sqrtf(ss), 1e-6f);

    const float ckv = ckO[b * 256 + t];
    float ss2 = bsum(ckv * ckv);
    nck[b * 256 + t] = ckv / fmaxf(sqrtf(ss2), 1e-6f);

    const float sg = signal[b * 256 + t];
    const float pv = previous[b * 256 + t];
    const float hd = hidden[b * 256 + t];
    const float s_mix = bsum(sg * W_mix[t] + pv * W_mix[256 + t]);
    const float s_wg  = bsum(sg * W_wg[t]  + hd * W_wg[256 + t]);
    const float s_pg  = bsum(sg * W_pg[t]  + hd * W_pg[256 + t]);
    const float s_mg  = bsum(sg * W_mg[t]  + hd * W_mg[256 + t]);
    if (t == 0) {
        gates[b * 8 + 0] = sigm(s_mix + b_mix[0]);   // read_mix
        gates[b * 8 + 1] = sigm(s_wg + b_wg[0]);     // write_strength
        gates[b * 8 + 2] = sigm(s_pg + b_pg[0]);     // persistence
        gates[b * 8 + 3] = s_mg + b_mg[0];           // merge gate pre-sigmoid
    }
    cread[b * 256 + t] = 0.f;
    pread[b * 256 + t] = 0.f;
}

// ---------------------------------------------------------------------------
// Kernel C: one pass over keys -> content score & similarity per (b,n).
// One wave per slot; float4 loads, shfl reduction. Keys become L2-resident.
// ---------------------------------------------------------------------------
__global__ __launch_bounds__(256) void scores(
    const float* __restrict__ keys, const float* __restrict__ q,
    const float* __restrict__ nck,
    float* __restrict__ csO, float* __restrict__ simO)
{
    const int b = blockIdx.y;
    const int n = blockIdx.x * 8 + (threadIdx.x >> 5);
    const int lane = threadIdx.x & 31;

    const float4* kp = (const float4*)(keys + (long)(b * NN + n) * 256);
    const float4* qp = (const float4*)(q + b * 256);
    const float4* cp = (const float4*)(nck + b * 256);
    const float4 k1 = kp[lane],      k2 = kp[32 + lane];
    const float4 q1 = qp[lane],      q2 = qp[32 + lane];
    const float4 c1 = cp[lane],      c2 = cp[32 + lane];

    float ss = d4(k1, k1) + d4(k2, k2);
    float cs = d4(q1, k1) + d4(q2, k2);
    float sv = d4(c1, k1) + d4(c2, k2);
    for (int off = 16; off > 0; off >>= 1) {
        ss += __shfl_down(ss, off, 32);
        cs += __shfl_down(cs, off, 32);
        sv += __shfl_down(sv, off, 32);
    }
    if (lane == 0) {
        const float inv = 1.f / fmaxf(sqrtf(ss), 1e-6f);
        csO[b * NN + n]  = cs * inv;
        simO[b * NN + n] = sv * inv;
    }
}

// ---------------------------------------------------------------------------
// Kernel D: per-batch reductions (softmax max/sum x2, top-3 similarity,
// replace argmax) + all scalar decision logic -> 16-float stats record.
// stats layout: [0]=maxc [1]=1/Zc [2]=maxp [3]=1/Zp [4]=key_mix [5]=value_mix
//               [6]=strength_boost_coef [7]=read_mix [8..10]=ow0..2
//               [12..14]=idx0..2 (int bits)
// ---------------------------------------------------------------------------
__global__ __launch_bounds__(256) void reduce_stats(
    const float* __restrict__ csA, const float* __restrict__ simA,
    const float* __restrict__ strength, const float* __restrict__ age,
    const float* __restrict__ gates, float* __restrict__ stats)
{
    __shared__ float sm[256];
    __shared__ float tv[768]; __shared__ int ti[768];
    __shared__ float rv[256]; __shared__ int ri[256];
    const int b = blockIdx.x, t = threadIdx.x;

    float mc = -1e30f, mp = -1e30f;
    float t0 = -1e30f, t1 = -1e30f, t2 = -1e30f;
    int i0 = -1, i1 = -1, i2 = -1;
    float br = -1e30f; int bi = 0;

    for (int j = t; j < NN; j += 256) {
        const float cs = csA[b * NN + j];
        const float st = strength[b * NN + j];
        const float ag = age[b * NN + j];
        const float sv = simA[b * NN + j];
        mc = fmaxf(mc, cs);
        mp = fmaxf(mp, 2.4f * st + 0.6f * (1.f - ag));
        if (sv > t0)      { t2=t1; i2=i1; t1=t0; i1=i0; t0=sv; i0=j; }
        else if (sv > t1) { t2=t1; i2=i1; t1=sv; i1=j; }
        else if (sv > t2) { t2=sv; i2=j; }
        const float rs = 1.2f * ag + (1.f - st) + 0.5f * (1.f - sv);
        if (rs > br) { br = rs; bi = j; }
    }

    sm[t] = mc; __syncthreads();
    for (int s = 128; s > 0; s >>= 1) { if (t < s) sm[t] = fmaxf(sm[t], sm[t + s]); __syncthreads(); }
    const float maxc = sm[0]; __syncthreads();
    sm[t] = mp; __syncthreads();
    for (int s = 128; s > 0; s >>= 1) { if (t < s) sm[t] = fmaxf(sm[t], sm[t + s]); __syncthreads(); }
    const float maxp = sm[0]; __syncthreads();

    tv[t*3] = t0; tv[t*3+1] = t1; tv[t*3+2] = t2;
    ti[t*3] = i0; ti[t*3+1] = i1; ti[t*3+2] = i2;
    rv[t] = br; ri[t] = bi;
    __syncthreads();

    float sc = 0.f, sp = 0.f;
    for (int j = t; j < NN; j += 256) {
        sc += expf((csA[b * NN + j] - maxc) * 2.f);
        const float st = strength[b * NN + j];
        const float ag = age[b * NN + j];
        sp += expf((2.4f * st + 0.6f * (1.f - ag) - maxp) * 2.f);
    }
    sm[t] = sc; __syncthreads();
    for (int s = 128; s > 0; s >>= 1) { if (t < s) sm[t] += sm[t + s]; __syncthreads(); }
    const float Zc = sm[0]; __syncthreads();
    sm[t] = sp; __syncthreads();
    for (int s = 128; s > 0; s >>= 1) { if (t < s) sm[t] += sm[t + s]; __syncthreads(); }
    const float Zp = sm[0]; __syncthreads();

    if (t == 0) {
        // merge top-3 across threads (value desc, ties -> lower index, like top_k)
        float T0=-1e30f, T1=-1e30f, T2=-1e30f; int I0=-1, I1=-1, I2=-1;
        for (int e = 0; e < 768; e++) {
            const float v = tv[e]; const int ii = ti[e];
            if (ii < 0) continue;
            if (v > T0 || (v == T0 && ii < I0)) { T2=T1; I2=I1; T1=T0; I1=I0; T0=v; I0=ii; }
            else if (v > T1 || (v == T1 && ii < I1)) { T2=T1; I2=I1; T1=v; I1=ii; }
            else if (v > T2 || (v == T2 && ii < I2)) { T2=v; I2=ii; }
        }
        float R = -1e30f; int RI = 0x7fffffff;
        for (int e = 0; e < 256; e++)
            if (rv[e] > R || (rv[e] == R && ri[e] < RI)) { R = rv[e]; RI = ri[e]; }

        const float rmix = gates[b*8+0], wst = gates[b*8+1];
        const float pers = gates[b*8+2], mgl = gates[b*8+3];
        const float maxs = T0, sec = T1;
        const float nov = fminf(fmaxf(1.f - maxs, 0.f), 1.f);
        const float mpref = sigm(mgl + 2.6f * maxs);
        const bool full    = (maxs > 0.78f) && (mpref >= 0.55f);
        const bool multi   = full && (sec > 0.68f);
        const bool partial = (!multi) && (maxs > 0.64f) && (sec > 0.52f);
        const float oscale = multi ? (0.16f + 0.52f * wst)
                           : partial ? (0.18f + 0.62f * wst)
                                     : (0.20f + 0.80f * wst);
        const float owf = oscale * (0.55f + 0.45f * nov);

        float ow0 = 0.f, ow1 = 0.f, ow2 = 0.f;
        int j0 = -1, j1 = -1, j2 = -1;
        if (multi) {
            const float e1 = expf(2.f * (T1 - T0));
            const float e2 = expf(2.f * (T2 - T0));
            const float z = 1.f + e1 + e2;
            j0 = I0; j1 = I1; j2 = I2;
            ow0 = owf / z; ow1 = owf * e1 / z; ow2 = owf * e2 / z;
        } else if (partial) {
            const float e1 = expf(2.f * (T1 - T0));
            const float z = 1.f + e1;
            j0 = I0; j1 = I1;
            ow0 = owf / z; ow1 = owf * e1 / z;
        } else if (full) {
            j0 = I0; ow0 = owf;
        } else {
            j0 = RI; ow0 = owf;
        }
        const bool ml = full || partial || multi;
        const float kmix = ml ? (0.28f + 0.24f * pers) : (0.78f + 0.16f * pers);
        const float vmix = ml ? (0.42f + 0.28f * pers) : (0.82f + 0.12f * pers);
        const float sb = 0.45f + 0.35f * pers + 0.45f * nov + 0.25f * wst;

        float* S = stats + b * 16;
        S[0] = maxc; S[1] = 1.f / Zc; S[2] = maxp; S[3] = 1.f / Zp;
        S[4] = kmix; S[5] = vmix; S[6] = sb; S[7] = rmix;
        S[8] = ow0;  S[9] = ow1;  S[10] = ow2; S[11] = 0.f;
        S[12] = __int_as_float(j0); S[13] = __int_as_float(j1);
        S[14] = __int_as_float(j2); S[15] = 0.f;
    }
}

// ---------------------------------------------------------------------------
// Kernel E: main streaming pass, b128 vectorized. 64 slots per block.
// Thread t owns dim-quad dq=t&63 and slot-group g=t>>6 (4 groups x 16 slots).
// keys: regular (L2-resident) loads; values: non-temporal (single use);
// all outputs: non-temporal stores (written once, never re-read).
// Weighted content/persistent sums accumulate in v4f registers, combined
// across slot-groups in LDS, flushed with one f32 atomicAdd per element.
// ---------------------------------------------------------------------------
__global__ __launch_bounds__(256) void update(
    const float* __restrict__ keys, const float* __restrict__ values,
    const float* __restrict__ strength, const float* __restrict__ age,
    const float* __restrict__ csA, const float* __restrict__ stats,
    const float* __restrict__ ck, const float* __restrict__ cv,
    float* __restrict__ outK, float* __restrict__ outV,
    float* __restrict__ outS, float* __restrict__ outA,
    float* __restrict__ cread, float* __restrict__ pread)
{
    __shared__ float swc[64], swp[64], sow[64], sst[16];
    __shared__ v4f red[256];
    const int b = blockIdx.y;
    const int n0 = blockIdx.x * 64;
    const int t = threadIdx.x;

    if (t < 16) sst[t] = stats[b * 16 + t];
    __syncthreads();
    const float maxc = sst[0], rZc = sst[1], maxp = sst[2], rZp = sst[3];
    const float kmix = sst[4], vmix = sst[5], sb = sst[6];
    const float ow0 = sst[8], ow1 = sst[9], ow2 = sst[10];
    const int j0 = __float_as_int(sst[12]);
    const int j1 = __float_as_int(sst[13]);
    const int j2 = __float_as_int(sst[14]);

    if (t < 64) {
        const int n = n0 + t;
        const long o = (long)b * NN + n;
        const float st = strength[o], ag = age[o], cs = csA[o];
        const float wc = expf((cs - maxc) * 2.f) * rZc;
        const float ps = 2.4f * st + 0.6f * (1.f - ag);
        const float wp = expf((ps - maxp) * 2.f) * rZp;
        const float ow = (n == j0) ? ow0 : (n == j1) ? ow1 : (n == j2) ? ow2 : 0.f;
        swc[t] = wc; swp[t] = wp; sow[t] = ow;
        __builtin_nontemporal_store(fminf(fmaxf(st * 0.995f + ow * sb, 0.f), 1.f), &outS[o]);
        __builtin_nontemporal_store(fminf(fmaxf((ag + 0.01f) * (1.f - ow), 0.f), 1.f), &outA[o]);
    }
    __syncthreads();

    const int g  = t >> 6;    // slot group 0..3
    const int dq = t & 63;    // dim quad: components dq*4 .. dq*4+3

    const v4f* kp = (const v4f*)keys;
    const v4f* vp = (const v4f*)values;
    v4f* oKp = (v4f*)outK;
    v4f* oVp = (v4f*)outV;
    const v4f ckd = ((const v4f*)(ck + b * 256))[dq];
    const v4f cvd = ((const v4f*)(cv + b * 256))[dq];

    v4f ca = {0.f, 0.f, 0.f, 0.f};
    v4f pa = {0.f, 0.f, 0.f, 0.f};
    const long base4 = ((long)b * NN + n0) * 64 + dq;   // float4 index
    for (int s = g; s < 64; s += 4) {
        const long idx4 = base4 + (long)s * 64;
        const v4f kk = kp[idx4];
        const v4f vv = __builtin_nontemporal_load(&vp[idx4]);
        const float ow = sow[s];
        const v4f ok = kk + (ow * kmix) * (ckd - kk);
        const v4f ov = vv + (ow * vmix) * (cvd - vv);
        __builtin_nontemporal_store(ok, &oKp[idx4]);
        __builtin_nontemporal_store(ov, &oVp[idx4]);
        ca += swc[s] * vv;
        pa += swp[s] * vv;
    }

    red[t] = ca;
    __syncthreads();
    if (t < 64) {
        const v4f s4 = red[t] + red[t + 64] + red[t + 128] + red[t + 192];
        float* dst = cread + b * 256 + dq * 4;
        atomicAdd(&dst[0], s4.x); atomicAdd(&dst[1], s4.y);
        atomicAdd(&dst[2], s4.z); atomicAdd(&dst[3], s4.w);
    }
    __syncthreads();
    red[t] = pa;
    __syncthreads();
    if (t < 64) {
        const v4f s4 = red[t] + red[t + 64] + red[t + 128] + red[t + 192];
        float* dst = pread + b * 256 + dq * 4;
        atomicAdd(&dst[0], s4.x); atomicAdd(&dst[1], s4.y);
        atomicAdd(&dst[2], s4.z); atomicAdd(&dst[3], s4.w);
    }
}

// ---------------------------------------------------------------------------
// Kernel F: memory_read = read_mix*content + (1-read_mix)*persistent
// ---------------------------------------------------------------------------
__global__ __launch_bounds__(256) void finish(
    const float* __restrict__ cread, const float* __restrict__ pread,
    const float* __restrict__ stats, float* __restrict__ mr)
{
    const int b = blockIdx.x, t = threadIdx.x;
    const float rmix = stats[b * 16 + 7];
    mr[b * 256 + t] = rmix * cread[b * 256 + t] + (1.f - rmix) * pread[b * 256 + t];
}

extern "C" void kernel_launch(void* const* d_in, const int* in_sizes, int n_in,
                              void* d_out, int out_size, void* d_ws, size_t ws_size,
                              hipStream_t stream) {
    (void)in_sizes; (void)n_in; (void)out_size; (void)ws_size;
    const float* signal   = (const float*)d_in[0];
    const float* previous = (const float*)d_in[1];
    const float* hidden   = (const float*)d_in[2];
    const float* keys     = (const float*)d_in[3];
    const float* values   = (const float*)d_in[4];
    const float* strength = (const float*)d_in[5];
    const float* age      = (const float*)d_in[6];
    const float* W_rq  = (const float*)d_in[7];
    const float* W_mix = (const float*)d_in[8];
    const float* b_mix = (const float*)d_in[9];
    const float* W_key = (const float*)d_in[10];
    const float* b_key = (const float*)d_in[11];
    const float* W_val = (const float*)d_in[12];
    const float* b_val = (const float*)d_in[13];
    const float* W_wg  = (const float*)d_in[14];
    const float* b_wg  = (const float*)d_in[15];
    const float* W_pg  = (const float*)d_in[16];
    const float* b_pg  = (const float*)d_in[17];
    const float* W_mg  = (const float*)d_in[18];
    const float* b_mg  = (const float*)d_in[19];

    float* w = (float*)d_ws;
    float* rq_raw = w;                 // 16384
    float* ck     = w + 16384;         // 16384
    float* cv     = w + 32768;         // 16384
    float* q      = w + 49152;         // 16384
    float* nck    = w + 65536;         // 16384
    float* gates  = w + 81920;         // 512
    float* cs     = w + 82432;         // 131072
    float* sim    = w + 213504;        // 131072
    float* stats  = w + 344576;        // 1024
    float* cread  = w + 345600;        // 16384
    float* pread  = w + 361984;        // 16384

    float* out  = (float*)d_out;
    float* mr   = out;                               // 16384
    float* outK = out + 16384;                       // 33554432
    float* outV = outK + (long)BB * NN * DD;         // 33554432
    float* outS = outV + (long)BB * NN * DD;         // 131072
    float* outA = outS + (long)BB * NN;              // 131072

    mm_wmma<<<dim3(16, 4, 3), 32, 0, stream>>>(
        signal, previous, hidden, W_rq, W_key, b_key, W_val, b_val,
        rq_raw, ck, cv);
    prep<<<BB, 256, 0, stream>>>(
        signal, previous, hidden, W_mix, b_mix, W_wg, b_wg, W_pg, b_pg,
        W_mg, b_mg, rq_raw, ck, q, nck, gates, cread, pread);
    scores<<<dim3(NN / 8, BB), 256, 0, stream>>>(keys, q, nck, cs, sim);
    reduce_stats<<<BB, 256, 0, stream>>>(cs, sim, strength, age, gates, stats);
    update<<<dim3(NN / 64, BB), 256, 0, stream>>>(
        keys, values, strength, age, cs, stats, ck, cv,
        outK, outV, outS, outA, cread, pread);
    finish<<<BB, 256, 0, stream>>>(cread, pread, stats, mr);
}